// LightGCN_33517924778087
// MI455X (gfx1250) — compile-verified
//
#include <hip/hip_runtime.h>
#include <hip/hip_bf16.h>

#define NUM_USER 100000
#define NUM_ITEM 50000
#define NODES    (NUM_USER + NUM_ITEM)
#define D        64
#define NNZ_E    5000000
#define BATCH_B  4096
#define ND       ((size_t)NODES * D)

typedef __attribute__((ext_vector_type(2))) float v2f;
typedef __attribute__((ext_vector_type(8))) float v8f;

// Concatenate user/item tables into e_a, seed acc, and zero e_b (first SpMM target).
__global__ void lgcn_init(const float* __restrict__ ut, const float* __restrict__ it,
                          float* __restrict__ e_a, float* __restrict__ acc,
                          float* __restrict__ e_b) {
    size_t i = (size_t)blockIdx.x * blockDim.x + threadIdx.x;
    if (i >= ND) return;
    float v = (i < (size_t)NUM_USER * D) ? ut[i] : it[i - (size_t)NUM_USER * D];
    e_a[i] = v;
    acc[i] = v;
    e_b[i] = 0.0f;
}

// Wave-per-edge SpMM: each lane owns 2 of the 64 dims.
// Gather x[col[e]] (b64 per lane, L2-resident: 38.4MB < 192MB L2),
// scale by vals[e], scatter with hardware f32 atomics.
__global__ void lgcn_spmm(const float* __restrict__ vals, const int* __restrict__ row,
                          const int* __restrict__ col, const float* __restrict__ x,
                          float* __restrict__ y) {
    unsigned tid  = blockIdx.x * blockDim.x + threadIdx.x;
    unsigned e    = tid >> 5;
    unsigned lane = tid & 31;
    if (e >= NNZ_E) return;
    int   r = row[e];
    int   c = col[e];
    float v = vals[e];
    const float2 xv = *(const float2*)(x + (size_t)c * D + lane * 2);
    float* dst = y + (size_t)r * D + lane * 2;
    unsafeAtomicAdd(dst,     v * xv.x);
    unsafeAtomicAdd(dst + 1, v * xv.y);
}

// acc += src; zero the other ping-pong buffer for the next SpMM.
__global__ void lgcn_addzero(float* __restrict__ acc, const float* __restrict__ src,
                             float* __restrict__ zbuf) {
    size_t i = (size_t)blockIdx.x * blockDim.x + threadIdx.x;
    if (i >= ND) return;
    acc[i] += src[i];
    zbuf[i] = 0.0f;
}

// WMMA scoring: one wave computes a 16x16 tile C = U(16x64) * I^T(64x16) via
// 16 chained V_WMMA_F32_16X16X4_F32, then extracts the diagonal through LDS.
// A 16x4 f32 layout: lane L -> row M=L&15, K = {2*(L>>4), 2*(L>>4)+1}.
// B 4x16 f32 layout mirrors it with N=L&15.  C: lane L, vgpr v -> M=v+8*(L>=16), N=L&15.
// out = acc/4 for each side -> divide products by 16.
__global__ void lgcn_score(const float* __restrict__ acc, const int* __restrict__ users,
                           const int* __restrict__ items, float* __restrict__ out) {
    __shared__ float tile[16 * 17];
    int lane = threadIdx.x;            // 0..31, one wave per block
    int b0   = blockIdx.x * 16;
    int j    = lane & 15;
    int kk0  = (lane >> 4) * 2;
    const float* urow = acc + (size_t)users[b0 + j] * D;
    const float* irow = acc + ((size_t)NUM_USER + (size_t)items[b0 + j]) * D;

    v8f c = {};
#pragma unroll
    for (int k = 0; k < 16; ++k) {
        v2f a, b;
        a.x = urow[4 * k + kk0];
        a.y = urow[4 * k + kk0 + 1];
        b.x = irow[4 * k + kk0];
        b.y = irow[4 * k + kk0 + 1];
        // 8 args: (neg_a, A, neg_b, B, c_mod, C, reuse_a, reuse_b)
        c = __builtin_amdgcn_wmma_f32_16x16x4_f32(false, a, false, b,
                                                  (short)0, c, false, false);
    }

    int mbase = (lane >> 4) * 8;
#pragma unroll
    for (int v = 0; v < 8; ++v)
        tile[(mbase + v) * 17 + j] = c[v];
    __syncthreads();

    if (lane < 16)
        out[b0 + lane] = tile[lane * 17 + lane] * (1.0f / 16.0f);
}

extern "C" void kernel_launch(void* const* d_in, const int* in_sizes, int n_in,
                              void* d_out, int out_size, void* d_ws, size_t ws_size,
                              hipStream_t stream) {
    const float* vals  = (const float*)d_in[0];
    const float* ut    = (const float*)d_in[1];
    const float* it    = (const float*)d_in[2];
    const int*   row   = (const int*)d_in[3];
    const int*   col   = (const int*)d_in[4];
    const int*   users = (const int*)d_in[5];
    const int*   items = (const int*)d_in[6];
    float*       out   = (float*)d_out;

    float* e_a = (float*)d_ws;
    float* e_b = e_a + ND;
    float* acc = e_b + ND;

    const int tpb  = 256;
    const int nblk = (int)((ND + tpb - 1) / tpb);
    const int sblk = (int)(((long long)NNZ_E * 32 + tpb - 1) / tpb);

    lgcn_init<<<nblk, tpb, 0, stream>>>(ut, it, e_a, acc, e_b);

    // layer 0: e_a -> e_b
    lgcn_spmm<<<sblk, tpb, 0, stream>>>(vals, row, col, e_a, e_b);
    lgcn_addzero<<<nblk, tpb, 0, stream>>>(acc, e_b, e_a);
    // layer 1: e_b -> e_a
    lgcn_spmm<<<sblk, tpb, 0, stream>>>(vals, row, col, e_b, e_a);
    lgcn_addzero<<<nblk, tpb, 0, stream>>>(acc, e_a, e_b);
    // layer 2: e_a -> e_b
    lgcn_spmm<<<sblk, tpb, 0, stream>>>(vals, row, col, e_a, e_b);
    lgcn_addzero<<<nblk, tpb, 0, stream>>>(acc, e_b, e_a);

    lgcn_score<<<BATCH_B / 16, 32, 0, stream>>>(acc, users, items, out);
}